// UpsampleOp_68118181315071
// MI455X (gfx1250) — compile-verified
//
#include <hip/hip_runtime.h>
#include <hip/hip_bf16.h>

// out[b,c,i,j] = x[b,c, srow(i), scol(j)]  -- pure parity gather.
// i even: srow = 2*((i/2)/3)+1   ; i odd: srow = 2*(((i+3)/2)/3)
// j even: scol = 2*((j/2)/3)+1   ; j odd: scol = 2*(((j+3)/2)/3)

#define PLANE_FLOATS (128 * 128)   // one (b,c) input plane = 64 KB, fits in LDS
#define OUT_HW 380                 // output plane is 380x380; 380 = 4*95

typedef int v4i_t __attribute__((ext_vector_type(4)));
typedef __attribute__((address_space(1))) v4i_t* gas_v4i_p;  // global
typedef __attribute__((address_space(3))) v4i_t* las_v4i_p;  // LDS

__device__ __forceinline__ unsigned lds_byte_off(const void* p) {
  // generic -> LDS(as3) addrspacecast, then ptrtoint: yields 32-bit LDS offset
  return (unsigned)(__SIZE_TYPE__)(__attribute__((address_space(3))) const void*)p;
}

__global__ __launch_bounds__(256) void upsample380_kernel(
    const float* __restrict__ x, float* __restrict__ out) {
  __shared__ float lds[PLANE_FLOATS];

  const int bc      = blockIdx.x;   // 0..127  (b*16 + c)
  const int quarter = blockIdx.y;   // 0..3    (95 output rows each)
  const int t       = threadIdx.x;  // 0..255  (8 wave32s)

  // ---------------- Phase 1: async DMA stage 64 KB plane into LDS ----------
  const float4* __restrict__ g4 =
      reinterpret_cast<const float4*>(x + (size_t)bc * PLANE_FLOATS);

#pragma unroll
  for (int k = 0; k < 16; ++k) {
    const int idx4 = k * 256 + t;   // 4096 float4 transfers per block
#if __has_builtin(__builtin_amdgcn_global_load_async_to_lds_b128)
    __builtin_amdgcn_global_load_async_to_lds_b128(
        (gas_v4i_p)(__SIZE_TYPE__)(g4 + idx4),
        (las_v4i_p)(__attribute__((address_space(3))) void*)(&lds[idx4 * 4]),
        0 /*imm offset*/, 0 /*cpol*/);
#else
    asm volatile("global_load_async_to_lds_b128 %0, %1, off"
                 :
                 : "v"(lds_byte_off(&lds[idx4 * 4])), "v"(g4 + idx4)
                 : "memory");
#endif
  }
#if __has_builtin(__builtin_amdgcn_s_wait_asynccnt)
  __builtin_amdgcn_s_wait_asynccnt(0);
#else
  asm volatile("s_wait_asynccnt 0" ::: "memory");
#endif
  __syncthreads();

  // ---------------- Phase 2: gather from LDS, coalesced b128 stores --------
  const int i0 = quarter * 95;
  float* __restrict__ op = out + (size_t)bc * (OUT_HW * OUT_HW);

  for (int item = t; item < 95 * 95; item += 256) {
    const int rl = item / 95;          // local output row 0..94
    const int j4 = item - rl * 95;     // float4 column group 0..94
    const int i  = i0 + rl;            // global output row

    unsigned srow;
    if (i & 1) srow = 2u * (((unsigned)(i + 3) >> 1) / 3u);
    else       srow = 2u * (((unsigned)i >> 1) / 3u) + 1u;

    const float* src = &lds[srow * 128];

    // columns j = 4*j4 .. 4*j4+3 ; q = j/2 for the even ones
    const unsigned q  = 2u * (unsigned)j4;
    const unsigned c0 = 2u * ( q       / 3u) + 1u;  // j even
    const unsigned c1 = 2u * ((q + 2u) / 3u);       // j odd
    const unsigned c2 = 2u * ((q + 1u) / 3u) + 1u;  // j even
    const unsigned c3 = 2u * ((q + 3u) / 3u);       // j odd

    float4 v;
    v.x = src[c0];
    v.y = src[c1];
    v.z = src[c2];
    v.w = src[c3];

    // row base is 16B aligned (380*4 = 1520 = 95*16), j4*4 keeps alignment
    *reinterpret_cast<float4*>(op + (size_t)i * OUT_HW + (j4 << 2)) = v;
  }
}

extern "C" void kernel_launch(void* const* d_in, const int* in_sizes, int n_in,
                              void* d_out, int out_size, void* d_ws, size_t ws_size,
                              hipStream_t stream) {
  (void)in_sizes; (void)n_in; (void)d_ws; (void)ws_size; (void)out_size;
  const float* x = (const float*)d_in[0];
  float* out = (float*)d_out;

  dim3 grid(128, 4, 1);   // 8*16 channels x 4 row-quarters
  dim3 block(256, 1, 1);  // 8 wave32s
  hipLaunchKernelGGL(upsample380_kernel, grid, block, 0, stream, x, out);
}